// WN_LVC_31971736551433
// MI455X (gfx1250) — compile-verified
//
#include <hip/hip_runtime.h>

#define BATCH 2
#define INCH 8
#define AUXC 80
#define DEPTH 8
#define DILC 64
#define RESC 64
#define SKIPC 64
#define PRED_H 32
#define PRED_LN 2
#define RADIX 3
#define STEPS 80
#define OFST 256
#define TLEN (STEPS*OFST)            // 20480
#define OUT_PER_G (2*DILC*RESC*RADIX) // 24576
#define BN_EPS 1e-5f

typedef __attribute__((ext_vector_type(2))) float v2f;
typedef __attribute__((ext_vector_type(8))) float v8f;

#if __has_builtin(__builtin_amdgcn_global_load_async_to_lds_b32)
#define USE_ASYNC_LDS 1
typedef __attribute__((address_space(1))) int gint;   // global (AS1) int
typedef __attribute__((address_space(3))) int lint;   // LDS (AS3) int
__device__ __forceinline__ void wait_async0(){
#if __has_builtin(__builtin_amdgcn_s_wait_asynccnt)
  __builtin_amdgcn_s_wait_asynccnt(0);
#else
  asm volatile("s_wait_asynccnt 0x0" ::: "memory");
#endif
}
#else
#define USE_ASYNC_LDS 0
#endif

// ---------------------------------------------------------------- utilities
__global__ void k_zero(float* __restrict__ p, int n){
  int i = blockIdx.x*blockDim.x + threadIdx.x;
  if(i<n) p[i]=0.f;
}

// ------------------------------------------------- start conv: (RES x IN) @ x
__global__ void k_start(const float* __restrict__ x, const float* __restrict__ w,
                        const float* __restrict__ bias, float* __restrict__ h){
  int i = blockIdx.x*blockDim.x + threadIdx.x;
  int n = BATCH*RESC*TLEN;
  if(i>=n) return;
  int t = i % TLEN; int o = (i/TLEN)%RESC; int b = i/(TLEN*RESC);
  float acc = bias[o];
  #pragma unroll
  for(int c=0;c<INCH;c++) acc += w[o*INCH+c]*x[((size_t)b*INCH+c)*TLEN+t];
  h[i]=acc;
}

// ------------------------------------------- predictor start: (256 x 80) @ y
__global__ void k_pred_start(const float* __restrict__ y, const float* __restrict__ w,
                             const float* __restrict__ bias, float* __restrict__ hp){
  int i = blockIdx.x*blockDim.x + threadIdx.x;
  int n = BATCH*256*STEPS;
  if(i>=n) return;
  int s = i%STEPS; int o = (i/STEPS)%256; int b = i/(STEPS*256);
  float acc = bias[o];
  const float* wr = w + o*AUXC;
  const float* yb = y + (size_t)b*AUXC*STEPS + s;
  #pragma unroll 8
  for(int c=0;c<AUXC;c++) acc += wr[c]*yb[c*STEPS];
  hp[i]=acc;
}

// ------------------- BN over (b,s) per channel + tanh (+ optional residual)
__global__ void k_bn_tanh(const float* __restrict__ in, const float* __restrict__ res,
                          float* __restrict__ out, const float* __restrict__ gamma,
                          const float* __restrict__ beta){
  __shared__ float sh[160];
  __shared__ float sh2[160];
  int ch = blockIdx.x;
  int t  = threadIdx.x;                 // 0..159
  int b = t/STEPS, s = t%STEPS;
  float x = in[((size_t)b*256+ch)*STEPS+s];
  sh[t]=x; sh2[t]=x*x;
  __syncthreads();
  for(int off=128; off>0; off>>=1){
    if(t<off && t+off<160){ sh[t]+=sh[t+off]; sh2[t]+=sh2[t+off]; }
    __syncthreads();
  }
  float mean = sh[0]*(1.f/160.f);
  float var  = sh2[0]*(1.f/160.f) - mean*mean;
  float sc = gamma[ch]*rsqrtf(var+BN_EPS);
  float val = tanhf((x-mean)*sc + beta[ch]);
  if(res) val += res[((size_t)b*256+ch)*STEPS+s];
  out[((size_t)b*256+ch)*STEPS+s]=val;
}

// ---------------------------------------- predictor grouped conv (8 groups)
__global__ void k_pgconv(const float* __restrict__ in, const float* __restrict__ w,
                         const float* __restrict__ bias, float* __restrict__ out){
  int i = blockIdx.x*blockDim.x + threadIdx.x;
  int n = BATCH*256*STEPS;
  if(i>=n) return;
  int s = i%STEPS; int ch=(i/STEPS)%256; int b=i/(STEPS*256);
  int g = ch/PRED_H, o = ch%PRED_H;
  float acc = bias[g*PRED_H+o];
  const float* wrow = w + (g*PRED_H+o)*PRED_H;
  const float* irow = in + ((size_t)b*256 + g*PRED_H)*STEPS + s;
  #pragma unroll
  for(int c=0;c<PRED_H;c++) acc += wrow[c]*irow[c*STEPS];
  out[i]=acc;
}

// -------- predictor end gconv for one depth; writes wts in (bs, r, o, c) order
__global__ void k_pred_end(const float* __restrict__ hp, const float* __restrict__ pew,
                           const float* __restrict__ peb, float* __restrict__ wts, int depth){
  int i = blockIdx.x*blockDim.x + threadIdx.x;
  const int n = BATCH*STEPS*RADIX*128*64;   // 3,932,160
  if(i>=n) return;
  int c = i & 63;
  int o = (i>>6) & 127;
  int rem = i >> 13;                        // bs*3 + r
  int r = rem % 3;
  int bs = rem / 3;
  int b = bs/STEPS, s = bs%STEPS;
  int ofull = o*(RESC*RADIX) + c*RADIX + r; // reference flattening o*192+c*3+r
  const float* wrow = pew + (size_t)(depth*OUT_PER_G + ofull)*PRED_H;
  const float* hrow = hp + ((size_t)b*256 + depth*PRED_H)*STEPS + s;
  float acc = peb[depth*OUT_PER_G + ofull];
  #pragma unroll
  for(int k=0;k<PRED_H;k++) acc += wrow[k]*hrow[k*STEPS];
  wts[i]=acc;
}

// -------------------- LVC layer GEMM via V_WMMA_F32_16X16X4_F32 (wave32)
// 1 workgroup (8 waves) per (b,s): z[128 x 256] = W[128 x (64*3)] @ Xg[(64*3) x 256]
// LDS window is stored channel-pair interleaved: hwin[pair][u][2] so that a
// B-fragment (two adjacent K rows at one time index) is a single ds_load_b64.
__global__ __launch_bounds__(256) void k_lvc(const float* __restrict__ h,
                                             const float* __restrict__ wts,
                                             float* __restrict__ z, int d){
  __shared__ float hwin[16*512];            // 16-channel chunk, pair-interleaved
  const int winW = OFST + 2*d;              // <= 512
  const int bs = blockIdx.x;
  const int b = bs/STEPS, s = bs%STEPS;
  const int tid = threadIdx.x;
  const int wave = tid>>5;
  const int lane = tid&31;
  const int lane16 = lane&15;
  const int koff = lane>>4;                 // 0: K0/K1 rows, 1: K2/K3 rows
  const int mo = wave;                      // M-strip 0..7
  const int t0glob = s*OFST - d;
  const float* wbase = wts + (size_t)bs*(RADIX*128*64);

  for(int npass=0; npass<2; ++npass){
    v8f acc[8] = {};
    for(int cc=0; cc<4; ++cc){
      __syncthreads();                      // previous chunk fully consumed
      for(int idx=tid; idx<16*winW; idx+=256){
        int cl = idx / winW;
        int u  = idx - cl*winW;
        int gt = t0glob + u;
        int ldsoff = ((cl>>1)*winW + u)*2 + (cl&1);  // pair-interleaved
#if USE_ASYNC_LDS
        if(gt>=0 && gt<TLEN){
          const float* src = h + ((size_t)b*RESC + cc*16 + cl)*TLEN + gt;
          __builtin_amdgcn_global_load_async_to_lds_b32(
              (gint*)(unsigned long long)src,
              (lint*)(unsigned)(unsigned long long)(hwin + ldsoff),
              0, 0);
        } else {
          hwin[ldsoff] = 0.f;
        }
#else
        float v = 0.f;
        if(gt>=0 && gt<TLEN) v = h[((size_t)b*RESC + cc*16 + cl)*TLEN + gt];
        hwin[ldsoff] = v;
#endif
      }
#if USE_ASYNC_LDS
      wait_async0();
#endif
      __syncthreads();
      for(int r=0;r<RADIX;r++){
        const float* wr = wbase + r*(128*64);
        #pragma unroll
        for(int c4=0;c4<4;c4++){
          int cg  = cc*16 + c4*4;           // global channel base of this K-quad
          int cl0 = c4*4;                   // local channel base in LDS
          // A fragment: lanes0-15 hold K0/K1, lanes16-31 hold K2/K3
          v2f a = *(const v2f*)(wr + (mo*16+lane16)*64 + cg + 2*koff);
          const float* brow = hwin + ((cl0>>1) + koff)*(winW*2);
          #pragma unroll
          for(int j=0;j<8;j++){
            int u = (npass*8+j)*16 + lane16 + r*d;
            v2f bf = *(const v2f*)(brow + u*2);   // single ds_load_b64
            acc[j] = __builtin_amdgcn_wmma_f32_16x16x4_f32(
                false, a, false, bf, (short)0, acc[j], false, false);
          }
        }
      }
    }
    float* zout = z + (size_t)bs*128*256;
    #pragma unroll
    for(int j=0;j<8;j++){
      int nn = (npass*8+j)*16 + lane16;
      #pragma unroll
      for(int v=0; v<8; v++){
        int m = mo*16 + v + 8*koff;         // D layout: M = v (+8 upper half)
        zout[m*256 + nn] = acc[j][v];
      }
    }
  }
}

// ----------------------------------------- gated activation tanh(zw)*sig(zv)
__global__ void k_gate(const float* __restrict__ z, float* __restrict__ g){
  int i = blockIdx.x*blockDim.x + threadIdx.x;
  int n = BATCH*RESC*TLEN;
  if(i>=n) return;
  int tg = i % TLEN; int c = (i/TLEN)%RESC; int b = i/(TLEN*RESC);
  int s = tg>>8; int t = tg&255;
  const float* zb = z + ((size_t)(b*STEPS+s)*128)*256;
  float zw = zb[c*256+t];
  float zv = zb[(c+64)*256+t];
  g[i] = tanhf(zw) * (1.f/(1.f+expf(-zv)));
}

// ----------- wo 1x1 conv GEMM (WMMA), fused residual/skip accumulation
// grid = (TLEN/16, M/16, B), 1 wave per 16x16 tile, K=64
__global__ __launch_bounds__(32) void k_wo(const float* __restrict__ g,
                                           const float* __restrict__ w,
                                           const float* __restrict__ bias,
                                           float* __restrict__ h, float* __restrict__ skip,
                                           int last){
  int n0 = blockIdx.x*16;
  int m0 = blockIdx.y*16;
  int b  = blockIdx.z;
  int lane = threadIdx.x;
  int lane16 = lane&15, koff = lane>>4;
  v8f acc = {};
  const float* gb = g + (size_t)b*RESC*TLEN;
  #pragma unroll
  for(int c0=0;c0<64;c0+=4){
    v2f a = *(const v2f*)(w + (m0+lane16)*64 + c0 + 2*koff);
    v2f bf;
    bf.x = gb[(size_t)(c0+2*koff    )*TLEN + n0 + lane16];
    bf.y = gb[(size_t)(c0+2*koff + 1)*TLEN + n0 + lane16];
    acc = __builtin_amdgcn_wmma_f32_16x16x4_f32(
        false, a, false, bf, (short)0, acc, false, false);
  }
  #pragma unroll
  for(int v=0; v<8; v++){
    int m = m0 + v + 8*koff;
    int nn = n0 + lane16;
    float val = acc[v] + bias[m];
    if(!last && m < 64){
      h[((size_t)b*RESC + m)*TLEN + nn] += val;          // residual into h
    } else {
      int sm = last ? m : (m-64);
      skip[((size_t)b*SKIPC + sm)*TLEN + nn] += val;     // accumulate skip
    }
  }
}

// ------------------------------------------------- end conv + output split
__global__ void k_end(const float* __restrict__ skip, const float* __restrict__ w,
                      const float* __restrict__ bias, float* __restrict__ out){
  int i = blockIdx.x*blockDim.x + threadIdx.x;
  int n = BATCH*2*INCH*TLEN;
  if(i>=n) return;
  int t = i%TLEN; int o = (i/TLEN)%(2*INCH); int b = i/(TLEN*2*INCH);
  float acc = bias[o];
  const float* wr = w + o*SKIPC;
  const float* sb = skip + (size_t)b*SKIPC*TLEN;
  #pragma unroll 8
  for(int c=0;c<SKIPC;c++) acc += wr[c]*sb[(size_t)c*TLEN+t];
  size_t dst;
  if(o<INCH) dst = ((size_t)b*INCH + o)*TLEN + t;                                  // tuple[0]
  else       dst = (size_t)BATCH*INCH*TLEN + ((size_t)b*INCH + (o-INCH))*TLEN + t; // tuple[1]
  out[dst]=acc;
}

extern "C" void kernel_launch(void* const* d_in, const int* in_sizes, int n_in,
                              void* d_out, int out_size, void* d_ws, size_t ws_size,
                              hipStream_t stream){
  const float* x     = (const float*)d_in[0];
  const float* y     = (const float*)d_in[1];
  const float* sw    = (const float*)d_in[2];
  const float* sb    = (const float*)d_in[3];
  const float* psw   = (const float*)d_in[4];
  const float* psb   = (const float*)d_in[5];
  const float* bn0g  = (const float*)d_in[6];
  const float* bn0b  = (const float*)d_in[7];
  const float* prw   = (const float*)d_in[8];
  const float* prb   = (const float*)d_in[9];
  const float* pbg   = (const float*)d_in[10];
  const float* pbb   = (const float*)d_in[11];
  const float* pew   = (const float*)d_in[12];
  const float* peb   = (const float*)d_in[13];
  const float* wo_w  = (const float*)d_in[14];
  const float* wo_b  = (const float*)d_in[15];
  const float* wol_w = (const float*)d_in[16];
  const float* wol_b = (const float*)d_in[17];
  const float* ew    = (const float*)d_in[18];
  const float* eb    = (const float*)d_in[19];

  float* ws   = (float*)d_ws;
  float* h    = ws;                 // B*64*T  = 2,621,440
  float* skip = ws + 2621440;       // B*64*T
  float* g    = ws + 5242880;       // B*64*T
  float* hp   = ws + 7864320;       // B*256*80 = 40,960
  float* hp2  = ws + 7905280;
  float* hp3  = ws + 7946240;
  float* wts  = ws + 7987200;       // B*S*3*128*64 = 3,932,160
  float* z    = ws + 11919360;      // B*S*128*256  = 5,242,880

  k_zero<<<(2621440+255)/256,256,0,stream>>>(skip, 2621440);
  k_start<<<(BATCH*RESC*TLEN+255)/256,256,0,stream>>>(x, sw, sb, h);

  // ---- kernel predictor ----
  k_pred_start<<<(BATCH*256*STEPS+255)/256,256,0,stream>>>(y, psw, psb, hp);
  k_bn_tanh<<<256,160,0,stream>>>(hp, nullptr, hp, bn0g, bn0b);
  for(int l=0;l<PRED_LN;l++){
    const float* rw0 = prw + (size_t)(l*2+0)*DEPTH*PRED_H*PRED_H;
    const float* rb0 = prb + (size_t)(l*2+0)*DEPTH*PRED_H;
    const float* rw1 = prw + (size_t)(l*2+1)*DEPTH*PRED_H*PRED_H;
    const float* rb1 = prb + (size_t)(l*2+1)*DEPTH*PRED_H;
    k_pgconv<<<(BATCH*256*STEPS+255)/256,256,0,stream>>>(hp, rw0, rb0, hp2);
    k_bn_tanh<<<256,160,0,stream>>>(hp2, nullptr, hp2, pbg+(l*2+0)*256, pbb+(l*2+0)*256);
    k_pgconv<<<(BATCH*256*STEPS+255)/256,256,0,stream>>>(hp2, rw1, rb1, hp3);
    k_bn_tanh<<<256,160,0,stream>>>(hp3, hp, hp, pbg+(l*2+1)*256, pbb+(l*2+1)*256);
  }

  // ---- LVC stack ----
  for(int i=0;i<DEPTH;i++){
    int d = 1<<i;
    int last = (i==DEPTH-1);
    k_pred_end<<<(3932160+255)/256,256,0,stream>>>(hp, pew, peb, wts, i);
    k_lvc<<<BATCH*STEPS,256,0,stream>>>(h, wts, z, d);
    k_gate<<<(BATCH*RESC*TLEN+255)/256,256,0,stream>>>(z, g);
    const float* w  = last ? wol_w : (wo_w + (size_t)i*(RESC+SKIPC)*DILC);
    const float* bb = last ? wol_b : (wo_b + (size_t)i*(RESC+SKIPC));
    int M = last ? 64 : 128;
    dim3 grid(TLEN/16, M/16, BATCH);
    k_wo<<<grid,32,0,stream>>>(g, w, bb, h, skip, last);
  }

  k_end<<<(BATCH*2*INCH*TLEN+255)/256,256,0,stream>>>(skip, ew, eb, (float*)d_out);
}